// NemotronHMOE_11364483465231
// MI455X (gfx1250) — compile-verified
//
#include <hip/hip_runtime.h>
#include <hip/hip_bf16.h>

// ---------------- model constants ----------------
#define T_   2048
#define D_   2048
#define DL_  1024
#define H_   512
#define SH_  2048
#define E_   64
#define K_   6
#define G_   8
#define TOPKG_ 4
#define C_   512
#define SCALE_ 2.5f

typedef __attribute__((ext_vector_type(2))) float v2f;
typedef __attribute__((ext_vector_type(8))) float v8f;

// --------------------------------------------------------------------
// gfx1250 async memory->LDS copy (GLOBAL_LOAD_ASYNC_TO_LDS_B128,
// tracked by ASYNCcnt). Builtin takes int4* in global (AS1) and
// LDS (AS3) address spaces. Falls back to a synchronous copy if the
// builtin is unavailable on this toolchain.
// --------------------------------------------------------------------
#if __has_builtin(__builtin_amdgcn_global_load_async_to_lds_b128)
#define HAVE_ASYNC_LDS 1
#else
#define HAVE_ASYNC_LDS 0
#endif

typedef __attribute__((__vector_size__(4 * sizeof(int)))) int vi4_t;
typedef __attribute__((address_space(1))) vi4_t gvi4_t;   // global int4
typedef __attribute__((address_space(3))) vi4_t lvi4_t;   // LDS int4

__device__ __forceinline__ void async_copy16(const float* g, float* l)
{
#if HAVE_ASYNC_LDS
    __builtin_amdgcn_global_load_async_to_lds_b128((gvi4_t*)g, (lvi4_t*)l, 0, 0);
#else
    *(float4*)l = *(const float4*)g;
#endif
}

__device__ __forceinline__ void wait_async_copies()
{
#if HAVE_ASYNC_LDS
#if __has_builtin(__builtin_amdgcn_s_wait_asynccnt)
    __builtin_amdgcn_s_wait_asynccnt(0);
#else
    asm volatile("s_wait_asynccnt 0x0" ::: "memory");
#endif
#endif
}

// =====================================================================
// Generic fp32 WMMA GEMM:  C[M,N] = act( A[M,K] * W[N,K]^T )  (+= if ACC)
// Block tile 128x64, K-chunk 16, 8 waves (256 thr), wave tile 32x32
// (4 x V_WMMA_F32_16X16X4_F32 accumulators, A/B fragment reuse).
// LDS double-buffered; tiles fetched with async-to-LDS DMA.
// =====================================================================
template <int ACT, int ACC>
__global__ __launch_bounds__(256) void gemm_nt(
    const float* __restrict__ A, const float* __restrict__ W,
    float* __restrict__ Cp, int M, int N, int Kd,
    long strideA, long strideW, long strideC)
{
    constexpr int BM = 128, BN = 64, BK = 16, BKP = 20; // pad 16->20 floats (80B, b128-aligned)
    __shared__ __align__(16) float As[2][BM * BKP];
    __shared__ __align__(16) float Ws[2][BN * BKP];

    const int tid  = threadIdx.x;
    const int lane = tid & 31;
    const int wid  = tid >> 5;     // 0..7
    const int wm   = wid & 3;      // 4 waves along M (32 rows each)
    const int wn   = wid >> 2;     // 2 waves along N (32 cols each)

    const int bm = blockIdx.y * BM;
    const int bn = blockIdx.x * BN;
    A  += (long)blockIdx.z * strideA;
    W  += (long)blockIdx.z * strideW;
    Cp += (long)blockIdx.z * strideC;

    // cooperative tile loaders: thread -> (row, 16B column group)
    const int lr = tid >> 2;         // 0..63
    const int lc = (tid & 3) * 4;    // 0,4,8,12
    const float* ga0 = A + (long)(bm + lr)      * Kd + lc;   // A rows 0..63
    const float* ga1 = A + (long)(bm + 64 + lr) * Kd + lc;   // A rows 64..127
    const float* gw  = W + (long)(bn + lr)      * Kd + lc;   // W rows 0..63
    const int la0 = lr * BKP + lc;
    const int la1 = (64 + lr) * BKP + lc;
    const int lw  = lr * BKP + lc;

    const int mrow  = lane & 15;           // fragment row/col index
    const int khalf = (lane >> 4) * 2;     // 0 or 2 (K offset for this half-wave)

    v8f acc00 = {0.f,0.f,0.f,0.f,0.f,0.f,0.f,0.f};
    v8f acc01 = acc00, acc10 = acc00, acc11 = acc00;

    // prologue: start DMA of chunk 0 into buffer 0
    async_copy16(ga0, &As[0][la0]);
    async_copy16(ga1, &As[0][la1]);
    async_copy16(gw,  &Ws[0][lw]);

    for (int k0 = 0; k0 < Kd; k0 += BK) {
        const int cur = (k0 / BK) & 1;
        wait_async_copies();           // this wave's pending DMA (current chunk) done
        __syncthreads();               // all waves' DMA visible; prev compute finished

        if (k0 + BK < Kd) {            // kick off next chunk while we compute
            const int nxt = cur ^ 1;
            async_copy16(ga0 + k0 + BK, &As[nxt][la0]);
            async_copy16(ga1 + k0 + BK, &As[nxt][la1]);
            async_copy16(gw  + k0 + BK, &Ws[nxt][lw]);
        }

        const float* as = As[cur];
        const float* ws = Ws[cur];
#pragma unroll
        for (int kk = 0; kk < BK; kk += 4) {
            const int kl = kk + khalf;
            // A fragments: rows wm*32 + {0,16} + mrow, K = kl..kl+1
            v2f a0 = *(const v2f*)&as[(wm * 32 +      mrow) * BKP + kl];
            v2f a1 = *(const v2f*)&as[(wm * 32 + 16 + mrow) * BKP + kl];
            // B fragments: B[k][n] = W[n][k]; n = wn*32 + {0,16} + mrow
            v2f b0 = *(const v2f*)&ws[(wn * 32 +      mrow) * BKP + kl];
            v2f b1 = *(const v2f*)&ws[(wn * 32 + 16 + mrow) * BKP + kl];
            acc00 = __builtin_amdgcn_wmma_f32_16x16x4_f32(false, a0, false, b0,
                                                          (short)0, acc00, false, false);
            acc01 = __builtin_amdgcn_wmma_f32_16x16x4_f32(false, a0, false, b1,
                                                          (short)0, acc01, false, false);
            acc10 = __builtin_amdgcn_wmma_f32_16x16x4_f32(false, a1, false, b0,
                                                          (short)0, acc10, false, false);
            acc11 = __builtin_amdgcn_wmma_f32_16x16x4_f32(false, a1, false, b1,
                                                          (short)0, acc11, false, false);
        }
    }

    // D layout: VGPR r -> row r (lanes 0-15) / row 8+r (lanes 16-31), col = lane&15
    const int sm0 = (lane >> 4) * 8;
    const int cn  = lane & 15;
#pragma unroll
    for (int i = 0; i < 2; ++i) {
#pragma unroll
        for (int j = 0; j < 2; ++j) {
            v8f acc = i ? (j ? acc11 : acc10) : (j ? acc01 : acc00);
            const int gn = bn + wn * 32 + j * 16 + cn;
#pragma unroll
            for (int r = 0; r < 8; ++r) {
                const int gm = bm + wm * 32 + i * 16 + sm0 + r;
                float v = acc[r];
                if (ACT == 1) { v = fmaxf(v, 0.f); v = v * v; }   // relu^2
                float* dst = Cp + (long)gm * N + gn;
                if (ACC) *dst = *dst + v; else *dst = v;
            }
        }
    }
}

// =====================================================================
// DeepseekV3 "noaux_tc" router: one thread per token.
// =====================================================================
__global__ void routing_kernel(const float* __restrict__ logits,
                               const float* __restrict__ bias,
                               int* __restrict__ topk_e,
                               float* __restrict__ tw)
{
    const int t = blockIdx.x * blockDim.x + threadIdx.x;
    if (t >= T_) return;

    float sc[E_], sfc[E_];
#pragma unroll 8
    for (int e = 0; e < E_; ++e) {
        float l = logits[t * E_ + e];
        float s = 1.0f / (1.0f + __expf(-l));
        sc[e]  = s;
        sfc[e] = s + bias[e];
    }
    // per-group sum of top-2
    float gs[G_];
    for (int g = 0; g < G_; ++g) {
        float m1 = -1e30f, m2 = -1e30f;
        for (int j = 0; j < E_ / G_; ++j) {
            float v = sfc[g * (E_ / G_) + j];
            if (v > m1) { m2 = m1; m1 = v; } else if (v > m2) { m2 = v; }
        }
        gs[g] = m1 + m2;
    }
    // top-4 groups (stable: strict > keeps lowest index on ties)
    unsigned gmask = 0;
    for (int i = 0; i < TOPKG_; ++i) {
        float best = -1e30f; int bg = 0;
        for (int g = 0; g < G_; ++g)
            if (!((gmask >> g) & 1u) && gs[g] > best) { best = gs[g]; bg = g; }
        gmask |= 1u << bg;
    }
    // top-6 experts over masked scores (non-selected groups -> 0.0, like reference)
    unsigned long long taken = 0ull;
    int   idx[K_]; float w[K_]; float sum = 0.f;
    for (int i = 0; i < K_; ++i) {
        float best = -1e30f; int be = 0;
        for (int e = 0; e < E_; ++e) {
            if ((taken >> e) & 1ull) continue;
            float v = ((gmask >> (e >> 3)) & 1u) ? sfc[e] : 0.0f;
            if (v > best) { best = v; be = e; }
        }
        taken |= 1ull << be;
        idx[i] = be; w[i] = sc[be]; sum += sc[be];
    }
    const float inv = SCALE_ / (sum + 1e-20f);
    for (int i = 0; i < K_; ++i) {
        topk_e[t * K_ + i] = idx[i];
        tw[t * K_ + i]     = w[i] * inv;
    }
}

// =====================================================================
// Capacity slot assignment: per-expert ordered prefix scan over the
// T*K flat assignment stream (matches reference cumsum semantics).
// grid = E blocks of 256 threads.
// =====================================================================
__global__ void scan_kernel(const int* __restrict__ topk_e, int* __restrict__ pos)
{
    const int e   = blockIdx.x;
    const int tid = threadIdx.x;
    __shared__ int s[256];
    int run = 0;
    for (int base = 0; base < T_ * K_; base += 256) {
        const int f = (topk_e[base + tid] == e) ? 1 : 0;
        s[tid] = f;
        __syncthreads();
        for (int off = 1; off < 256; off <<= 1) {
            int v = (tid >= off) ? s[tid - off] : 0;
            __syncthreads();
            s[tid] += v;
            __syncthreads();
        }
        if (f) pos[base + tid] = run + s[tid] - 1;
        run += s[255];
        __syncthreads();
    }
}

// =====================================================================
// Dispatch: buf[e, slot, :] = xl[token, :]   (skip capacity overflow)
// grid = T*K blocks, 256 threads, float4 copies (1024 floats per row).
// =====================================================================
__global__ void dispatch_kernel(const float* __restrict__ xl,
                                const int* __restrict__ topk_e,
                                const int* __restrict__ pos,
                                float* __restrict__ buf)
{
    const int a = blockIdx.x;
    const int p = pos[a];
    if (p >= C_) return;
    const int e   = topk_e[a];
    const int tok = a / K_;
    const float4* src = (const float4*)(xl + (size_t)tok * DL_);
    float4*       dst = (float4*)(buf + ((size_t)e * C_ + p) * DL_);
    dst[threadIdx.x] = src[threadIdx.x];
}

// =====================================================================
// Combine: routed_lat[t,:] = sum_k tw[t,k] * y[e, slot, :]
// grid = T blocks, 256 threads * float4 = 1024 = DL.
// =====================================================================
__global__ void combine_kernel(const float* __restrict__ y,
                               const int* __restrict__ topk_e,
                               const int* __restrict__ pos,
                               const float* __restrict__ tw,
                               float* __restrict__ rlat)
{
    const int t = blockIdx.x;
    const int d = threadIdx.x * 4;
    float4 acc = make_float4(0.f, 0.f, 0.f, 0.f);
#pragma unroll
    for (int k = 0; k < K_; ++k) {
        const int a = t * K_ + k;
        const int p = pos[a];
        if (p >= C_) continue;
        const int e   = topk_e[a];
        const float g = tw[a];
        const float4 v = *(const float4*)(y + ((size_t)e * C_ + p) * DL_ + d);
        acc.x += g * v.x; acc.y += g * v.y; acc.z += g * v.z; acc.w += g * v.w;
    }
    *(float4*)(rlat + (size_t)t * DL_ + d) = acc;
}

// =====================================================================
// Host launcher
// =====================================================================
extern "C" void kernel_launch(void* const* d_in, const int* in_sizes, int n_in,
                              void* d_out, int out_size, void* d_ws, size_t ws_size,
                              hipStream_t stream)
{
    const float* x      = (const float*)d_in[0];  // [T,D]
    const float* gate_w = (const float*)d_in[1];  // [E,D]
    const float* gate_b = (const float*)d_in[2];  // [E]
    const float* fc1_w  = (const float*)d_in[3];  // [DL,D]
    const float* fc2_w  = (const float*)d_in[4];  // [D,DL]
    const float* w1     = (const float*)d_in[5];  // [E,H,DL]
    const float* w2     = (const float*)d_in[6];  // [E,DL,H]
    const float* sup    = (const float*)d_in[7];  // [SH,D]
    const float* sdn    = (const float*)d_in[8];  // [D,SH]
    float* out = (float*)d_out;                   // [T,D]

    // ---- workspace layout (floats) ----
    float* ws = (float*)d_ws;
    size_t off = 0;
    float* logits = ws + off; off += (size_t)T_ * E_;
    int*   topk_e = (int*)(ws + off); off += (size_t)T_ * K_;
    float* tw     = ws + off; off += (size_t)T_ * K_;
    int*   pos    = (int*)(ws + off); off += (size_t)T_ * K_;
    float* xl     = ws + off; off += (size_t)T_ * DL_;
    float* rlat   = ws + off; off += (size_t)T_ * DL_;
    float* h      = ws + off; off += (size_t)E_ * C_ * H_;
    float* y      = ws + off; off += (size_t)E_ * C_ * DL_;
    float* buf    = ws + off; off += (size_t)E_ * C_ * DL_;
    float* s1     = buf;  // alias: s1 (shared-expert hidden) dead before buf is written

    const dim3 blk(256);

    // 1) router logits: [T,E] = x @ gate_w^T
    gemm_nt<0, 0><<<dim3(E_ / 64, T_ / 128, 1), blk, 0, stream>>>(
        x, gate_w, logits, T_, E_, D_, 0, 0, 0);

    // 2) routing (top-k with group limit)
    routing_kernel<<<T_ / 128, 128, 0, stream>>>(logits, gate_b, topk_e, tw);

    // 3) shared experts: s1 = relu2(x @ sup^T); out = s1 @ sdn^T
    gemm_nt<1, 0><<<dim3(SH_ / 64, T_ / 128, 1), blk, 0, stream>>>(
        x, sup, s1, T_, SH_, D_, 0, 0, 0);
    gemm_nt<0, 0><<<dim3(D_ / 64, T_ / 128, 1), blk, 0, stream>>>(
        s1, sdn, out, T_, D_, SH_, 0, 0, 0);

    // 4) latent down-projection: xl = x @ fc1^T
    gemm_nt<0, 0><<<dim3(DL_ / 64, T_ / 128, 1), blk, 0, stream>>>(
        x, fc1_w, xl, T_, DL_, D_, 0, 0, 0);

    // 5) capacity slots + dispatch
    scan_kernel<<<E_, 256, 0, stream>>>(topk_e, pos);
    dispatch_kernel<<<T_ * K_, 256, 0, stream>>>(xl, topk_e, pos, buf);

    // 6) grouped expert GEMMs (batched over grid.z = E)
    gemm_nt<1, 0><<<dim3(H_ / 64, C_ / 128, E_), blk, 0, stream>>>(
        buf, w1, h, C_, H_, DL_, (long)C_ * DL_, (long)H_ * DL_, (long)C_ * H_);
    gemm_nt<0, 0><<<dim3(DL_ / 64, C_ / 128, E_), blk, 0, stream>>>(
        h, w2, y, C_, DL_, H_, (long)C_ * H_, (long)DL_ * H_, (long)C_ * DL_);

    // 7) weighted combine to latent
    combine_kernel<<<T_, 256, 0, stream>>>(y, topk_e, pos, tw, rlat);

    // 8) latent up-projection, accumulate onto shared output
    gemm_nt<0, 1><<<dim3(D_ / 64, T_ / 128, 1), blk, 0, stream>>>(
        rlat, fc2_w, out, T_, D_, DL_, 0, 0, 0);
}